// Criterion_48773648613658
// MI455X (gfx1250) — compile-verified
//
#include <hip/hip_runtime.h>
#include <hip/hip_bf16.h>

// ---------------------------------------------------------------------------
// Criterion loss on MI455X (gfx1250, wave32).
//
// argmin_f ||a - c_f||^2 == argmin_f (-2 a.c_f + ||c_f||^2)
// Pairwise term computed with V_WMMA_F32_16X16X4_F32:
//   A row = [px,py,pz,1],  B col = [-2cx,-2cy,-2cz,|c|^2].
// Each wave holds 4 independent 16-row accumulator sets, sweeps all 1250 face
// tiles 2 at a time (8 WMMAs/iter) with B-panel loads pipelined one iteration
// ahead. Per-lane running min + face index; 4-step shfl_xor butterfly within
// each 16-lane half finalizes the per-row argmin.
// ---------------------------------------------------------------------------

typedef __attribute__((ext_vector_type(2))) float v2f;
typedef __attribute__((ext_vector_type(8))) float v8f;

#define N_CLOTH  20000
#define N_FACES  20000
#define F_TILES  (N_FACES / 16)              // 1250 (exact, even)
#define A_TILES  ((N_CLOTH + 15) / 16)       // 1250 (exact)
#define TPW      4                           // A-tiles per wave
#define NGROUPS  ((A_TILES + TPW - 1) / TPW) // 313
#define EPS_C    1e-3f
#define NPART    ((N_CLOTH + 255) / 256)     // 79 partial sums

// ---------------------------------------------------------------------------
// Phase 1: per-face prep.
//   bpack[f] = (-2cx, -2cy, -2cz, |c|^2)  from current obstacle positions
//   ncent[f] = next-step centroid, nnorm[f] = next-step unit normal
// ---------------------------------------------------------------------------
__global__ void __launch_bounds__(256)
crit_prep(const float* __restrict__ obs_pos,
          const float* __restrict__ obs_next,
          const int*   __restrict__ faces,
          float4* __restrict__ bpack,
          float4* __restrict__ ncent,
          float4* __restrict__ nnorm)
{
    int f = blockIdx.x * blockDim.x + threadIdx.x;
    if (f >= N_FACES) return;
    int i0 = faces[3 * f + 0];
    int i1 = faces[3 * f + 1];
    int i2 = faces[3 * f + 2];

    // current centroid -> WMMA B-panel entry
    float ax = obs_pos[3 * i0 + 0], ay = obs_pos[3 * i0 + 1], az = obs_pos[3 * i0 + 2];
    float bx = obs_pos[3 * i1 + 0], by = obs_pos[3 * i1 + 1], bz = obs_pos[3 * i1 + 2];
    float cx = obs_pos[3 * i2 + 0], cy = obs_pos[3 * i2 + 1], cz = obs_pos[3 * i2 + 2];
    const float k3 = 1.0f / 3.0f;
    float gx = (ax + bx + cx) * k3;
    float gy = (ay + by + cy) * k3;
    float gz = (az + bz + cz) * k3;
    bpack[f] = make_float4(-2.0f * gx, -2.0f * gy, -2.0f * gz,
                           gx * gx + gy * gy + gz * gz);

    // next-step centroid + unit normal
    float qax = obs_next[3 * i0 + 0], qay = obs_next[3 * i0 + 1], qaz = obs_next[3 * i0 + 2];
    float qbx = obs_next[3 * i1 + 0], qby = obs_next[3 * i1 + 1], qbz = obs_next[3 * i1 + 2];
    float qcx = obs_next[3 * i2 + 0], qcy = obs_next[3 * i2 + 1], qcz = obs_next[3 * i2 + 2];
    float mx = (qax + qbx + qcx) * k3;
    float my = (qay + qby + qcy) * k3;
    float mz = (qaz + qbz + qcz) * k3;
    float e1x = qbx - qax, e1y = qby - qay, e1z = qbz - qaz;
    float e2x = qcx - qax, e2y = qcy - qay, e2z = qcz - qaz;
    float nx = e1y * e2z - e1z * e2y;
    float ny = e1z * e2x - e1x * e2z;
    float nz = e1x * e2y - e1y * e2x;
    float len = sqrtf(nx * nx + ny * ny + nz * nz);
    float inv = 1.0f / fmaxf(len, 1e-12f);
    ncent[f] = make_float4(mx, my, mz, 0.0f);
    nnorm[f] = make_float4(nx * inv, ny * inv, nz * inv, 0.0f);
}

// ---------------------------------------------------------------------------
// Phase 2: nearest face per cloth node via V_WMMA_F32_16X16X4_F32.
// A layout (32-bit 16x4, ISA 7.12.2): lane m holds row M=m;
//   VGPR0 = K0 (lanes 0-15) / K2 (lanes 16-31), VGPR1 = K1 / K3.
// B layout mirrored: lane n holds col N=n; VGPR0 = K0/K2, VGPR1 = K1/K3.
// D layout: VGPR v = row v (lanes 0-15) / row v+8 (lanes 16-31), col = lane&15.
// ---------------------------------------------------------------------------
__global__ void __launch_bounds__(32)
crit_argmin(const float*  __restrict__ cloth_pos,
            const float4* __restrict__ bpack,
            int*          __restrict__ face_idx)
{
    const int lane = threadIdx.x & 31;
    const int half = lane >> 4;        // 0: K0/K1 + rows v, 1: K2/K3 + rows v+8
    const int m    = lane & 15;
    const int g    = blockIdx.x;

    // Load the 4 A-tiles (cloth rows), clamp to last tile if short.
    v2f a[TPW];
    int rowBase[TPW];
    #pragma unroll
    for (int j = 0; j < TPW; ++j) {
        int tile = g * TPW + j;
        if (tile >= A_TILES) tile = A_TILES - 1;
        rowBase[j] = tile * 16;
        int row = tile * 16 + m;       // < N_CLOTH (1250*16 == 20000)
        float px = cloth_pos[3 * row + 0];
        float py = cloth_pos[3 * row + 1];
        float pz = cloth_pos[3 * row + 2];
        a[j].x = half ? pz   : px;     // K2 : K0
        a[j].y = half ? 1.0f : py;     // K3 : K1  (1 multiplies |c|^2 row)
    }

    v8f minv[TPW];
    int idxv[TPW][8];
    #pragma unroll
    for (int j = 0; j < TPW; ++j)
        #pragma unroll
        for (int v = 0; v < 8; ++v) { minv[j][v] = 3.4e38f; idxv[j][v] = 0; }

    const v8f zacc = {};

    // Software-pipelined B-panel loads: fetch tiles t, t+1 one iteration ahead.
    // The final over-read lands in ncent (same workspace) and is discarded.
    float4 bp0 = bpack[m];
    float4 bp1 = bpack[16 + m];

    for (int t = 0; t < F_TILES; t += 2) {
        float4 nbp0 = bpack[(t + 2) * 16 + m];
        float4 nbp1 = bpack[(t + 3) * 16 + m];

        v2f b0, b1;
        b0.x = half ? bp0.z : bp0.x;   // -2cz  : -2cx
        b0.y = half ? bp0.w : bp0.y;   // |c|^2 : -2cy
        b1.x = half ? bp1.z : bp1.x;
        b1.y = half ? bp1.w : bp1.y;

        // 8 independent WMMAs -> no D->A/B RAW between them.
        v8f d0 = __builtin_amdgcn_wmma_f32_16x16x4_f32(false, a[0], false, b0, (short)0, zacc, false, false);
        v8f d1 = __builtin_amdgcn_wmma_f32_16x16x4_f32(false, a[1], false, b0, (short)0, zacc, false, false);
        v8f d2 = __builtin_amdgcn_wmma_f32_16x16x4_f32(false, a[2], false, b0, (short)0, zacc, false, false);
        v8f d3 = __builtin_amdgcn_wmma_f32_16x16x4_f32(false, a[3], false, b0, (short)0, zacc, false, false);
        v8f e0 = __builtin_amdgcn_wmma_f32_16x16x4_f32(false, a[0], false, b1, (short)0, zacc, false, false);
        v8f e1 = __builtin_amdgcn_wmma_f32_16x16x4_f32(false, a[1], false, b1, (short)0, zacc, false, false);
        v8f e2 = __builtin_amdgcn_wmma_f32_16x16x4_f32(false, a[2], false, b1, (short)0, zacc, false, false);
        v8f e3 = __builtin_amdgcn_wmma_f32_16x16x4_f32(false, a[3], false, b1, (short)0, zacc, false, false);

        const int f0 = t * 16 + m;     // face index this lane's column holds
        const int f1 = f0 + 16;
        #pragma unroll
        for (int v = 0; v < 8; ++v) {
            // tile t: strict < keeps the earliest (lowest) face index
            bool l0 = d0[v] < minv[0][v];
            minv[0][v] = l0 ? d0[v] : minv[0][v];
            idxv[0][v] = l0 ? f0    : idxv[0][v];
            bool l1 = d1[v] < minv[1][v];
            minv[1][v] = l1 ? d1[v] : minv[1][v];
            idxv[1][v] = l1 ? f0    : idxv[1][v];
            bool l2 = d2[v] < minv[2][v];
            minv[2][v] = l2 ? d2[v] : minv[2][v];
            idxv[2][v] = l2 ? f0    : idxv[2][v];
            bool l3 = d3[v] < minv[3][v];
            minv[3][v] = l3 ? d3[v] : minv[3][v];
            idxv[3][v] = l3 ? f0    : idxv[3][v];
            // tile t+1
            bool k0 = e0[v] < minv[0][v];
            minv[0][v] = k0 ? e0[v] : minv[0][v];
            idxv[0][v] = k0 ? f1    : idxv[0][v];
            bool k1 = e1[v] < minv[1][v];
            minv[1][v] = k1 ? e1[v] : minv[1][v];
            idxv[1][v] = k1 ? f1    : idxv[1][v];
            bool k2 = e2[v] < minv[2][v];
            minv[2][v] = k2 ? e2[v] : minv[2][v];
            idxv[2][v] = k2 ? f1    : idxv[2][v];
            bool k3s = e3[v] < minv[3][v];
            minv[3][v] = k3s ? e3[v] : minv[3][v];
            idxv[3][v] = k3s ? f1    : idxv[3][v];
        }

        bp0 = nbp0;
        bp1 = nbp1;
    }

    // Per-row argmin across the 16 lanes of each half (wave32 butterfly).
    #pragma unroll
    for (int j = 0; j < TPW; ++j) {
        #pragma unroll
        for (int v = 0; v < 8; ++v) {
            float val = minv[j][v];
            int   ix  = idxv[j][v];
            #pragma unroll
            for (int off = 1; off < 16; off <<= 1) {
                float oval = __shfl_xor(val, off, 32);
                int   oix  = __shfl_xor(ix,  off, 32);
                if (oval < val || (oval == val && oix < ix)) { val = oval; ix = oix; }
            }
            if (m == 0) {
                int row = rowBase[j] + v + 8 * half;
                face_idx[row] = ix;    // duplicate tiles rewrite identical data
            }
        }
    }
}

// ---------------------------------------------------------------------------
// Phase 3: cubic hinge penalty per node -> per-block partial sums.
// ---------------------------------------------------------------------------
__global__ void __launch_bounds__(256)
crit_loss(const float*  __restrict__ cloth_pred,
          const int*    __restrict__ face_idx,
          const float4* __restrict__ ncent,
          const float4* __restrict__ nnorm,
          float*        __restrict__ partials)
{
    __shared__ float sm[256];
    int i = blockIdx.x * 256 + threadIdx.x;
    float p = 0.0f;
    if (i < N_CLOTH) {
        int f = face_idx[i];
        float4 c = ncent[f];
        float4 n = nnorm[f];
        float dx = cloth_pred[3 * i + 0] - c.x;
        float dy = cloth_pred[3 * i + 1] - c.y;
        float dz = cloth_pred[3 * i + 2] - c.z;
        float dist = dx * n.x + dy * n.y + dz * n.z;
        float t = fmaxf(EPS_C - dist, 0.0f);
        p = t * t * t;
    }
    sm[threadIdx.x] = p;
    __syncthreads();
    #pragma unroll
    for (int s = 128; s > 0; s >>= 1) {
        if (threadIdx.x < s) sm[threadIdx.x] += sm[threadIdx.x + s];
        __syncthreads();
    }
    if (threadIdx.x == 0) partials[blockIdx.x] = sm[0];
}

__global__ void __launch_bounds__(256)
crit_finalize(const float* __restrict__ partials,
              const int*   __restrict__ iter_num,
              float*       __restrict__ out)
{
    __shared__ float sm[256];
    float s = 0.0f;
    for (int k = threadIdx.x; k < NPART; k += 256) s += partials[k];
    sm[threadIdx.x] = s;
    __syncthreads();
    #pragma unroll
    for (int st = 128; st > 0; st >>= 1) {
        if (threadIdx.x < st) sm[threadIdx.x] += sm[threadIdx.x + st];
        __syncthreads();
    }
    if (threadIdx.x == 0) {
        int it = iter_num[0];                 // START_RAMPUP == 0
        if (it < 0) it = 0;
        float progress = fminf((float)it / 50000.0f, 1.0f);
        float w = 1.0f + 999.0f * progress;   // WEIGHT_START..WEIGHT_MAX
        out[0] = w * sm[0];
    }
}

// ---------------------------------------------------------------------------
// Launch wrapper. Inputs (setup_inputs order):
//   0 cloth_pos[N,3] f32   1 cloth_pred_pos[N,3] f32
//   2 obstacle_pos[M,3] f32  3 obstacle_next_pos[M,3] f32
//   4 obstacle_faces[F,3] i32  5 iter_num i32 scalar
// Workspace layout (all 16B aligned):
//   [0, 320000)        bpack  float4[F]   (+512B over-read slack into ncent)
//   [320000, 640000)   ncent  float4[F]
//   [640000, 960000)   nnorm  float4[F]
//   [960000, 1040000)  face_idx int[N]
//   [1040000, ...)     partials float[NPART]
// ---------------------------------------------------------------------------
extern "C" void kernel_launch(void* const* d_in, const int* in_sizes, int n_in,
                              void* d_out, int out_size, void* d_ws, size_t ws_size,
                              hipStream_t stream)
{
    const float* cloth_pos  = (const float*)d_in[0];
    const float* cloth_pred = (const float*)d_in[1];
    const float* obs_pos    = (const float*)d_in[2];
    const float* obs_next   = (const float*)d_in[3];
    const int*   faces      = (const int*)d_in[4];
    const int*   iter_num   = (const int*)d_in[5];
    float*       out        = (float*)d_out;

    char* ws = (char*)d_ws;
    float4* bpack    = (float4*)(ws + 0);
    float4* ncent    = (float4*)(ws + 320000);
    float4* nnorm    = (float4*)(ws + 640000);
    int*    face_idx = (int*)   (ws + 960000);
    float*  partials = (float*) (ws + 1040000);

    crit_prep<<<(N_FACES + 255) / 256, 256, 0, stream>>>(
        obs_pos, obs_next, faces, bpack, ncent, nnorm);

    crit_argmin<<<NGROUPS, 32, 0, stream>>>(cloth_pos, bpack, face_idx);

    crit_loss<<<NPART, 256, 0, stream>>>(cloth_pred, face_idx, ncent, nnorm, partials);

    crit_finalize<<<1, 256, 0, stream>>>(partials, iter_num, out);
}